// Encoder_54202487275957
// MI455X (gfx1250) — compile-verified
//
#include <hip/hip_runtime.h>

typedef __attribute__((ext_vector_type(2))) float v2f;
typedef __attribute__((ext_vector_type(8))) float v8f;

#define F_IN  128
#define H_DIM 128
#define O_DIM 64

// ---------------------------------------------------------------------------
// init: deg = 1.0 (self-loop weight), zero the scatter accumulators
// ---------------------------------------------------------------------------
__global__ void k_init(float* __restrict__ deg, float* __restrict__ hacc,
                       float* __restrict__ out, int n, int hElems, int outElems) {
    int idx = blockIdx.x * blockDim.x + threadIdx.x;
    if (idx < n)        deg[idx]  = 1.0f;
    if (idx < hElems)   hacc[idx] = 0.0f;
    if (idx < outElems) out[idx]  = 0.0f;
}

// ---------------------------------------------------------------------------
// deg[col[e]] += w[e]   (self-loop "+1" already folded into init)
// ---------------------------------------------------------------------------
__global__ void k_deg(const long long* __restrict__ col,
                      const float* __restrict__ w,
                      float* __restrict__ deg, int E) {
    int e = blockIdx.x * blockDim.x + threadIdx.x;
    if (e < E) atomicAdd(&deg[(int)col[e]], w[e]);
}

// ---------------------------------------------------------------------------
// dinv = deg > 0 ? rsqrt(deg) : 0
// ---------------------------------------------------------------------------
__global__ void k_dinv(const float* __restrict__ deg, float* __restrict__ dinv, int n) {
    int i = blockIdx.x * blockDim.x + threadIdx.x;
    if (i < n) {
        float d = deg[i];
        dinv[i] = (d > 0.0f) ? __frsqrt_rn(d) : 0.0f;
    }
}

// ---------------------------------------------------------------------------
// Y[nRows, COUT] = X[nRows, K] @ W[K, COUT]  via V_WMMA_F32_16X16X4_F32.
// One wave computes one 16x16 output tile; K fully unrolled (32 WMMAs).
// A 16x4 frag: lanes 0-15 hold (K=k0,k0+1) of row m=lane; lanes 16-31 hold
// (K=k0+2,k0+3) of row m=lane-16.  B 4x16 frag mirrored over columns.
// C/D: VGPR j -> (M=j, N=lane) for lanes 0-15, (M=j+8, N=lane-16) for 16-31.
// ---------------------------------------------------------------------------
template <int K, int COUT>
__global__ void k_gemm_wmma(const float* __restrict__ X,
                            const float* __restrict__ W,
                            float* __restrict__ Y, int nRows) {
    const int lane   = threadIdx.x & 31;
    const int wid    = (blockIdx.x * blockDim.x + threadIdx.x) >> 5;
    const int tilesN = COUT / 16;
    const int nTiles = (nRows / 16) * tilesN;
    if (wid >= nTiles) return;

    const int tileM = wid / tilesN;
    const int tileN = wid % tilesN;

    const int mn    = lane & 15;          // row (for A) / col (for B) within tile
    const int khalf = (lane >> 4) << 1;   // 0 for lanes 0-15, 2 for lanes 16-31

    const float* __restrict__ xRow = X + (size_t)(tileM * 16 + mn) * K;
    const float* __restrict__ wCol = W + (size_t)(tileN * 16 + mn);

    v8f acc = {0.0f, 0.0f, 0.0f, 0.0f, 0.0f, 0.0f, 0.0f, 0.0f};

#pragma unroll
    for (int k0 = 0; k0 < K; k0 += 4) {
        v2f a, b;
        a.x = xRow[k0 + khalf];
        a.y = xRow[k0 + khalf + 1];
        b.x = wCol[(size_t)(k0 + khalf) * COUT];
        b.y = wCol[(size_t)(k0 + khalf + 1) * COUT];
        acc = __builtin_amdgcn_wmma_f32_16x16x4_f32(
            /*neg_a=*/false, a, /*neg_b=*/false, b,
            /*c_mod=*/(short)0, acc, /*reuse_a=*/false, /*reuse_b=*/false);
    }

    const int mOff = (lane >> 4) << 3;    // 0 or 8
    float* __restrict__ yBase = Y + (size_t)tileN * 16 + mn;
#pragma unroll
    for (int j = 0; j < 8; ++j) {
        yBase[(size_t)(tileM * 16 + j + mOff) * COUT] = acc[j];
    }
}

// ---------------------------------------------------------------------------
// Edge scatter: dst[c] += dinv[r]*w*dinv[c] * src[r]   (one wave-slice / edge,
// float4 channel chunks, f32 global atomics -> stays in L2)
// ---------------------------------------------------------------------------
__global__ void k_scatter(const long long* __restrict__ row,
                          const long long* __restrict__ col,
                          const float* __restrict__ w,
                          const float* __restrict__ dinv,
                          const float* __restrict__ src,
                          float* __restrict__ dst, int E, int C) {
    const int lanesPerEdge = C >> 2;                    // 32 (C=128) or 16 (C=64)
    int gid = blockIdx.x * blockDim.x + threadIdx.x;
    int e = gid / lanesPerEdge;
    if (e >= E) return;
    int ch = (gid % lanesPerEdge) << 2;

    int r = (int)row[e];
    int c = (int)col[e];
    float nrm = dinv[r] * w[e] * dinv[c];

    const float4 v = *reinterpret_cast<const float4*>(src + (size_t)r * C + ch);
    float* p = dst + (size_t)c * C + ch;
    atomicAdd(p + 0, nrm * v.x);
    atomicAdd(p + 1, nrm * v.y);
    atomicAdd(p + 2, nrm * v.z);
    atomicAdd(p + 3, nrm * v.w);
}

// ---------------------------------------------------------------------------
// Epilogue: dst += dinv[i]^2 * selfsrc (self-loop term, norm = dinv*1*dinv)
//           + bias, optional ReLU.
// ---------------------------------------------------------------------------
__global__ void k_epilogue(float* __restrict__ dst,
                           const float* __restrict__ selfsrc,
                           const float* __restrict__ dinv,
                           const float* __restrict__ bias,
                           int n, int C, int do_relu) {
    int idx = blockIdx.x * blockDim.x + threadIdx.x;
    if (idx >= n * C) return;
    int i  = idx / C;
    int ch = idx - i * C;
    float d = dinv[i];
    float v = dst[idx] + d * d * selfsrc[idx] + bias[ch];
    dst[idx] = do_relu ? fmaxf(v, 0.0f) : v;
}

// ---------------------------------------------------------------------------
extern "C" void kernel_launch(void* const* d_in, const int* in_sizes, int n_in,
                              void* d_out, int out_size, void* d_ws, size_t ws_size,
                              hipStream_t stream) {
    const float*     x   = (const float*)d_in[0];
    const long long* ei  = (const long long*)d_in[1];
    const float*     ew  = (const float*)d_in[2];
    const float*     W1  = (const float*)d_in[3];
    const float*     b1  = (const float*)d_in[4];
    const float*     Wmu = (const float*)d_in[5];
    const float*     bmu = (const float*)d_in[6];
    const float*     Wls = (const float*)d_in[7];
    const float*     bls = (const float*)d_in[8];

    const int n = in_sizes[0] / F_IN;     // 100000 (divisible by 16)
    const int E = in_sizes[2];            // 1600000
    const long long* rowIdx = ei;         // edge_index[0]
    const long long* colIdx = ei + E;     // edge_index[1]

    // workspace layout (floats): deg[n] | dinv[n] | xw[n*128] | hacc[n*128]
    // xw is reused as {tmu[n*64], tls[n*64]} after layer 1.
    float* ws   = (float*)d_ws;
    float* deg  = ws;
    float* dinv = deg + n;
    float* xw   = dinv + n;
    float* hacc = xw + (size_t)n * H_DIM;
    float* tmu  = xw;
    float* tls  = xw + (size_t)n * O_DIM;

    float* mu = (float*)d_out;
    float* ls = mu + (size_t)n * O_DIM;

    const int BS = 256;
    const int hElems   = n * H_DIM;           // 12.8M
    const int outElems = n * O_DIM * 2;       // 12.8M

    // 1) init: deg=1 (self-loops), zero hacc and d_out
    {
        int total = hElems > outElems ? hElems : outElems;
        k_init<<<(total + BS - 1) / BS, BS, 0, stream>>>(deg, hacc, (float*)d_out,
                                                         n, hElems, outElems);
    }
    // 2) degree accumulation over destination nodes
    k_deg<<<(E + BS - 1) / BS, BS, 0, stream>>>(colIdx, ew, deg, E);
    // 3) dinv = rsqrt(deg)
    k_dinv<<<(n + BS - 1) / BS, BS, 0, stream>>>(deg, dinv, n);

    // ---- Layer 1: h = relu(GCN(x, W1) + b1) ----
    {
        int tiles = (n / 16) * (H_DIM / 16);
        k_gemm_wmma<F_IN, H_DIM><<<(tiles * 32 + BS - 1) / BS, BS, 0, stream>>>(x, W1, xw, n);
        long long thr = (long long)E * (H_DIM / 4);
        k_scatter<<<(int)((thr + BS - 1) / BS), BS, 0, stream>>>(rowIdx, colIdx, ew, dinv,
                                                                 xw, hacc, E, H_DIM);
        k_epilogue<<<(hElems + BS - 1) / BS, BS, 0, stream>>>(hacc, xw, dinv, b1,
                                                              n, H_DIM, /*relu=*/1);
    }

    // ---- Layers 2 & 3: mu = GCN(h, Wmu)+bmu ; logstd = GCN(h, Wls)+bls ----
    {
        int tiles = (n / 16) * (O_DIM / 16);
        int gemmGrid = (tiles * 32 + BS - 1) / BS;
        k_gemm_wmma<H_DIM, O_DIM><<<gemmGrid, BS, 0, stream>>>(hacc, Wmu, tmu, n);
        k_gemm_wmma<H_DIM, O_DIM><<<gemmGrid, BS, 0, stream>>>(hacc, Wls, tls, n);

        long long thr = (long long)E * (O_DIM / 4);
        int scatGrid = (int)((thr + BS - 1) / BS);
        k_scatter<<<scatGrid, BS, 0, stream>>>(rowIdx, colIdx, ew, dinv, tmu, mu, E, O_DIM);
        k_scatter<<<scatGrid, BS, 0, stream>>>(rowIdx, colIdx, ew, dinv, tls, ls, E, O_DIM);

        int ne = n * O_DIM;
        k_epilogue<<<(ne + BS - 1) / BS, BS, 0, stream>>>(mu, tmu, dinv, bmu, n, O_DIM, 0);
        k_epilogue<<<(ne + BS - 1) / BS, BS, 0, stream>>>(ls, tls, dinv, bls, n, O_DIM, 0);
    }
}